// GATModel_50946902065603
// MI455X (gfx1250) — compile-verified
//
#include <hip/hip_runtime.h>
#include <math.h>

#define IN_CH 128
#define HID 64
#define OUT_CH 10
#define N_GRAPHS 64
#define NEG_SLOPE 0.2f
#define EPS_PYG 1e-16f

typedef float v2f __attribute__((ext_vector_type(2)));
typedef float v8f __attribute__((ext_vector_type(8)));

// ---------------------------------------------------------------- utilities
__global__ __launch_bounds__(256) void k_fill(float* __restrict__ p, float v, int n) {
    int i = blockIdx.x * blockDim.x + threadIdx.x;
    if (i < n) p[i] = v;
}

__device__ inline void atomic_max_f32(float* addr, float v) {
    // init value must be -inf. Signed-int max for v>=0, unsigned-int min for v<0.
    if (v >= 0.0f) atomicMax((int*)addr, __float_as_int(v));
    else           atomicMin((unsigned int*)addr, (unsigned int)__float_as_int(v));
}

// ------------------------------------------------- xl = x@Wl, xr = x@Wr (WMMA f32)
// One wave computes a 16x64 tile of both xl and xr using V_WMMA_F32_16X16X4_F32.
// A frag (16x4 f32): lanes 0-15 rows M=0..15 hold K={k0,k0+1}; lanes 16-31 hold K={k0+2,k0+3}.
// B frag (4x16 f32): lanes 0-15 cols N,  K={k0,k0+1}; lanes 16-31 K={k0+2,k0+3}.
// C/D (16x16 f32): VGPR r holds row r (lanes 0-15) / row r+8 (lanes 16-31).
// LDS holds W in K-PAIRED layout: sW[kp*HID+col] = {W[2kp,col], W[2kp+1,col]}
// so a B fragment is a single ds_load_b64 into an adjacent (even-aligned) VGPR pair.
__global__ __launch_bounds__(256) void k_gemm_xlxr(
    const float* __restrict__ x, const float* __restrict__ Wl,
    const float* __restrict__ Wr, float* __restrict__ xl,
    float* __restrict__ xr, int n_nodes)
{
    __shared__ float2 sW[2 * (IN_CH / 2) * HID];   // 64 KB: Wl pairs then Wr pairs
    const int tid = threadIdx.x;
    const int NPAIR = (IN_CH / 2) * HID;            // 4096 float2 per matrix
    for (int i = tid; i < NPAIR; i += 256) {
        int kp  = i / HID;
        int col = i - kp * HID;
        sW[i]         = make_float2(Wl[(2 * kp) * HID + col], Wl[(2 * kp + 1) * HID + col]);
        sW[NPAIR + i] = make_float2(Wr[(2 * kp) * HID + col], Wr[(2 * kp + 1) * HID + col]);
    }
    __syncthreads();

    const int wave = tid >> 5;
    const int lane = tid & 31;
    const int tile = blockIdx.x * 8 + wave;         // 16-node tile per wave
    const int n_tiles = (n_nodes + 15) >> 4;
    if (tile >= n_tiles) return;                    // wave-uniform: EXEC all-1s inside

    const int row0 = tile * 16;
    const int m    = lane & 15;
    const int kh   = lane >> 4;                     // lane-half selects K pair

    int rrow = row0 + m;                            // clamp (keeps EXEC full for WMMA)
    if (rrow > n_nodes - 1) rrow = n_nodes - 1;
    const float* xrow = x + (size_t)rrow * IN_CH;
    const float2* sWl = sW;
    const float2* sWr = sW + NPAIR;

    v8f accL[4] = {};
    v8f accR[4] = {};

    for (int k0 = 0; k0 < IN_CH; k0 += 4) {
        const int kp = (k0 >> 1) + kh;              // K-pair index for this lane half
        v2f a = *(const v2f*)(xrow + k0 + kh * 2);  // one b64 global load
        #pragma unroll
        for (int nt = 0; nt < 4; ++nt) {
            const int col = nt * 16 + m;
            v2f bl = *(const v2f*)&sWl[kp * HID + col];  // one ds_load_b64
            v2f br = *(const v2f*)&sWr[kp * HID + col];  // one ds_load_b64
            accL[nt] = __builtin_amdgcn_wmma_f32_16x16x4_f32(
                false, a, false, bl, (short)0, accL[nt], false, false);
            accR[nt] = __builtin_amdgcn_wmma_f32_16x16x4_f32(
                false, a, false, br, (short)0, accR[nt], false, false);
        }
    }

    // Store C/D. Fast path: whole 16-row tile in range (always true when N%16==0).
    float* pl = xl + (size_t)(row0 + 8 * kh) * HID + m;
    float* pr = xr + (size_t)(row0 + 8 * kh) * HID + m;
    if (row0 + 16 <= n_nodes) {
        #pragma unroll
        for (int r = 0; r < 8; ++r) {
            #pragma unroll
            for (int nt = 0; nt < 4; ++nt) {
                pl[(size_t)r * HID + nt * 16] = accL[nt][r];
                pr[(size_t)r * HID + nt * 16] = accR[nt][r];
            }
        }
    } else {
        #pragma unroll
        for (int r = 0; r < 8; ++r) {
            if (row0 + 8 * kh + r < n_nodes) {
                #pragma unroll
                for (int nt = 0; nt < 4; ++nt) {
                    pl[(size_t)r * HID + nt * 16] = accL[nt][r];
                    pr[(size_t)r * HID + nt * 16] = accR[nt][r];
                }
            }
        }
    }
}

// --------------------- e = leakyrelu(xl[src]+xr[tgt]) . att ; m[tgt] = max(m[tgt], e)
__global__ __launch_bounds__(256) void k_edge_score_max(
    const float* __restrict__ xl, const float* __restrict__ xr,
    const float* __restrict__ att, const int* __restrict__ src,
    const int* __restrict__ tgt, float* __restrict__ e,
    float* __restrict__ m, int n_edges)
{
    __shared__ float satt[HID];
    if (threadIdx.x < HID) satt[threadIdx.x] = att[threadIdx.x];
    __syncthreads();
    int i = blockIdx.x * blockDim.x + threadIdx.x;
    if (i >= n_edges) return;
    int s = src[i], t = tgt[i];
    const float4* pl = (const float4*)(xl + (size_t)s * HID);
    const float4* pr = (const float4*)(xr + (size_t)t * HID);
    // prefetch the xl[src] / xr[tgt] rows this thread gathers for its own edge
    // (also exercises global_prefetch_b8); harmless if OOB-speculative.
    __builtin_prefetch(pl, 0, 1);
    __builtin_prefetch(pr, 0, 1);
    float acc = 0.0f;
    #pragma unroll
    for (int h = 0; h < HID / 4; ++h) {
        float4 a4 = pl[h];
        float4 b4 = pr[h];
        float z;
        z = a4.x + b4.x; acc += satt[4 * h + 0] * (z > 0.0f ? z : NEG_SLOPE * z);
        z = a4.y + b4.y; acc += satt[4 * h + 1] * (z > 0.0f ? z : NEG_SLOPE * z);
        z = a4.z + b4.z; acc += satt[4 * h + 2] * (z > 0.0f ? z : NEG_SLOPE * z);
        z = a4.w + b4.w; acc += satt[4 * h + 3] * (z > 0.0f ? z : NEG_SLOPE * z);
    }
    e[i] = acc;
    atomic_max_f32(&m[t], acc);
}

// ------------------------------------------------- a = exp(e-m[tgt]); denom += a
__global__ __launch_bounds__(256) void k_exp_sum(
    const float* __restrict__ e, const float* __restrict__ m,
    const int* __restrict__ tgt, float* __restrict__ a,
    float* __restrict__ denom, int n_edges)
{
    int i = blockIdx.x * blockDim.x + threadIdx.x;
    if (i >= n_edges) return;
    int t = tgt[i];
    float v = __expf(e[i] - m[t]);
    a[i] = v;
    atomicAdd(&denom[t], v);
}

// ------------------------------------------------- out[tgt] += alpha * xl[src]
// 16 threads per edge, 4 channels each.
__global__ __launch_bounds__(256) void k_scatter(
    const float* __restrict__ xl, const float* __restrict__ a,
    const float* __restrict__ denom, const int* __restrict__ src,
    const int* __restrict__ tgt, float* __restrict__ out, int n_edges)
{
    int idx = blockIdx.x * blockDim.x + threadIdx.x;
    int ed = idx >> 4;
    if (ed >= n_edges) return;
    int cg = (idx & 15) * 4;
    int t = tgt[ed];
    float alpha = a[ed] / (denom[t] + EPS_PYG);
    const float4 v = *(const float4*)(xl + (size_t)src[ed] * HID + cg);
    float* o = out + (size_t)t * HID + cg;
    atomicAdd(o + 0, alpha * v.x);
    atomicAdd(o + 1, alpha * v.y);
    atomicAdd(o + 2, alpha * v.z);
    atomicAdd(o + 3, alpha * v.w);
}

// ------------------------------------------------- pooled[g] += out[n]+bias ; cnt[g]++
__global__ __launch_bounds__(256) void k_pool(
    const float* __restrict__ out, const float* __restrict__ bias,
    const int* __restrict__ batch, float* __restrict__ pooled,
    float* __restrict__ cnt, int n_nodes)
{
    int idx = blockIdx.x * blockDim.x + threadIdx.x;
    int node = idx >> 4;
    if (node >= n_nodes) return;
    int cg = (idx & 15) * 4;
    int g = batch[node];
    const float* po = out + (size_t)node * HID + cg;
    float* pp = pooled + g * HID + cg;
    atomicAdd(pp + 0, po[0] + bias[cg + 0]);
    atomicAdd(pp + 1, po[1] + bias[cg + 1]);
    atomicAdd(pp + 2, po[2] + bias[cg + 2]);
    atomicAdd(pp + 3, po[3] + bias[cg + 3]);
    if (cg == 0) atomicAdd(&cnt[g], 1.0f);
}

// ------------------------------------------------- y = leakyrelu(pooled/cnt) @ fc_w + fc_b
__global__ __launch_bounds__(640) void k_final(
    const float* __restrict__ pooled, const float* __restrict__ cnt,
    const float* __restrict__ fc_w, const float* __restrict__ fc_b,
    float* __restrict__ y)
{
    int i = threadIdx.x;
    if (i >= N_GRAPHS * OUT_CH) return;
    int g = i / OUT_CH, o = i % OUT_CH;
    float c = cnt[g]; if (c < 1.0f) c = 1.0f;
    float inv = 1.0f / c;
    float acc = fc_b[o];
    #pragma unroll 8
    for (int h = 0; h < HID; ++h) {
        float p = pooled[g * HID + h] * inv;
        p = (p > 0.0f) ? p : 0.01f * p;
        acc += p * fc_w[h * OUT_CH + o];
    }
    y[i] = acc;
}

// ---------------------------------------------------------------- launcher
extern "C" void kernel_launch(void* const* d_in, const int* in_sizes, int n_in,
                              void* d_out, int out_size, void* d_ws, size_t ws_size,
                              hipStream_t stream) {
    const float* x    = (const float*)d_in[0];
    const float* Wl   = (const float*)d_in[1];
    const float* Wr   = (const float*)d_in[2];
    const float* att  = (const float*)d_in[3];
    const float* bias = (const float*)d_in[4];
    const float* fc_w = (const float*)d_in[5];
    const float* fc_b = (const float*)d_in[6];
    const int*   ei   = (const int*)d_in[7];
    const int*   bat  = (const int*)d_in[8];

    const int n_nodes = in_sizes[0] / IN_CH;
    const int n_edges = in_sizes[7] / 2;
    const int* src = ei;
    const int* tgt = ei + n_edges;

    // scratch carve-out (256B aligned slabs)
    char* w = (char*)d_ws;
    size_t off = 0;
    #define TAKE(name, nfl) \
        float* name = (float*)(w + off); \
        off += (((size_t)(nfl) * sizeof(float) + 255) & ~(size_t)255);
    TAKE(xl,     (size_t)n_nodes * HID)
    TAKE(xr,     (size_t)n_nodes * HID)
    TAKE(outb,   (size_t)n_nodes * HID)
    TAKE(ebuf,   (size_t)n_edges)
    TAKE(abuf,   (size_t)n_edges)
    TAKE(mbuf,   (size_t)n_nodes)
    TAKE(dbuf,   (size_t)n_nodes)
    TAKE(pooled, (size_t)N_GRAPHS * HID)
    TAKE(cntb,   (size_t)N_GRAPHS)
    #undef TAKE
    (void)ws_size;

    const int B = 256;
    // init scratch accumulators
    k_fill<<<(n_nodes * HID + B - 1) / B, B, 0, stream>>>(outb, 0.0f, n_nodes * HID);
    k_fill<<<(n_nodes + B - 1) / B, B, 0, stream>>>(mbuf, -INFINITY, n_nodes);
    k_fill<<<(n_nodes + B - 1) / B, B, 0, stream>>>(dbuf, 0.0f, n_nodes);
    k_fill<<<(N_GRAPHS * HID + B - 1) / B, B, 0, stream>>>(pooled, 0.0f, N_GRAPHS * HID);
    k_fill<<<(N_GRAPHS + B - 1) / B, B, 0, stream>>>(cntb, 0.0f, N_GRAPHS);

    // xl/xr GEMMs via WMMA: 8 waves (16-node tiles) per block
    const int n_tiles = (n_nodes + 15) / 16;
    k_gemm_xlxr<<<(n_tiles + 7) / 8, B, 0, stream>>>(x, Wl, Wr, xl, xr, n_nodes);

    // edge phase (score + segment-max fused)
    k_edge_score_max<<<(n_edges + B - 1) / B, B, 0, stream>>>(xl, xr, att, src, tgt, ebuf, mbuf, n_edges);
    k_exp_sum<<<(n_edges + B - 1) / B, B, 0, stream>>>(ebuf, mbuf, tgt, abuf, dbuf, n_edges);
    {
        long long threads = (long long)n_edges * 16;
        k_scatter<<<(int)((threads + B - 1) / B), B, 0, stream>>>(xl, abuf, dbuf, src, tgt, outb, n_edges);
    }

    // pooling + head
    {
        long long threads = (long long)n_nodes * 16;
        k_pool<<<(int)((threads + B - 1) / B), B, 0, stream>>>(outb, bias, bat, pooled, cntb, n_nodes);
    }
    k_final<<<1, 640, 0, stream>>>(pooled, cntb, fc_w, fc_b, (float*)d_out);
}